// FeatureAttentionLayer_38912403701950
// MI455X (gfx1250) — compile-verified
//
#include <hip/hip_runtime.h>
#include <hip/hip_bf16.h>
#include <math.h>

#define ALPHA 0.2f
#define B_SZ 4
#define W_SZ 128
#define K_SZ 256
#define E_SZ 256
#define CHUNK 32   // e-rows of tT staged into LDS per async round (32KB)

typedef __attribute__((ext_vector_type(2))) float v2f;
typedef __attribute__((ext_vector_type(8))) float v8f;

// ---------------------------------------------------------------------------
// Kernel 1: u = v @ W1^T  and  tT = (v @ W2^T + lin_b) stored transposed.
//   v[b,k,w] = x[b,w,k].  One wave computes one 16x16 tile via
//   32x V_WMMA_F32_16X16X4_F32 (inner dim W=128 in steps of 4).
// grid: (32 [n-tile: 0-15 => u, 16-31 => t], 16 [m-tile over k], 4 [batch])
// ---------------------------------------------------------------------------
__global__ void __launch_bounds__(32)
gemm_ut_kernel(const float* __restrict__ x,      // (B, W, K)
               const float* __restrict__ lin_w,  // (E, 2W)
               const float* __restrict__ lin_b,  // (E)
               float* __restrict__ u,            // (B, K, E)
               float* __restrict__ tT)           // (B, E, K)
{
    const int lane = threadIdx.x;
    const int half = lane >> 4;      // 0: lanes 0-15, 1: lanes 16-31
    const int lm   = lane & 15;

    const bool is_t   = (blockIdx.x >= 16);
    const int  e_base = (blockIdx.x & 15) * 16;
    const int  k_base = blockIdx.y * 16;
    const int  b      = blockIdx.z;
    const int  w_off  = is_t ? W_SZ : 0;

    const float* xb = x + (size_t)b * W_SZ * K_SZ;   // xb[w*K + k]

    v8f acc = {};
    for (int w0 = 0; w0 < W_SZ; w0 += 4) {
        const int kk = w0 + 2 * half;
        // A 16x4 (f32): lane lm holds row m=lm; vgpr j holds K = 2*half + j
        v2f amat;
        amat.x = xb[(size_t)(kk + 0) * K_SZ + k_base + lm];
        amat.y = xb[(size_t)(kk + 1) * K_SZ + k_base + lm];
        // B 4x16 (f32): lane lm holds col n=lm; vgpr j holds row K = 2*half + j
        const float* wrow = lin_w + (size_t)(e_base + lm) * (2 * W_SZ) + w_off + kk;
        v2f bmat;
        bmat.x = wrow[0];
        bmat.y = wrow[1];
        acc = __builtin_amdgcn_wmma_f32_16x16x4_f32(
            false, amat, false, bmat, (short)0, acc, false, false);
    }

    // C/D layout: vgpr r -> (m = r + 8*half, n = lm)
    if (!is_t) {
        float* up = u + (size_t)b * K_SZ * E_SZ + (size_t)e_base + lm;
#pragma unroll
        for (int r = 0; r < 8; ++r) {
            const int m = r + 8 * half;
            up[(size_t)(k_base + m) * E_SZ] = acc[r];
        }
    } else {
        const float bb = lin_b[e_base + lm];
        float* tp = tT + (size_t)b * E_SZ * K_SZ + (size_t)(e_base + lm) * K_SZ + k_base;
#pragma unroll
        for (int r = 0; r < 8; ++r) {
            const int m = r + 8 * half;
            tp[m] = acc[r] + bb;
        }
    }
}

// ---------------------------------------------------------------------------
// Kernel 2: per (b,i) row: e[j] = sum_e a[e]*leakyrelu(u[i,e]+t[j,e]) + bias,
//           softmax over j, write attn row.  256 threads, thread j owns col j.
//   tT chunks are staged into LDS via GLOBAL_LOAD_ASYNC_TO_LDS_B128
//   (each 32-row chunk of tT is a contiguous 32KB block in memory).
// grid: (K, B), block: 256
// ---------------------------------------------------------------------------
__global__ void __launch_bounds__(256)
attn_row_kernel(const float* __restrict__ u,     // (B, K, E)
                const float* __restrict__ tT,    // (B, E, K)
                const float* __restrict__ a_vec, // (E)
                const float* __restrict__ bias,  // (K, K)
                float* __restrict__ attn)        // (B, K, K)
{
    __shared__ float t_sh[CHUNK * K_SZ];   // 32 KB staging tile
    __shared__ float u_sh[E_SZ];
    __shared__ float a_sh[E_SZ];
    __shared__ float red[K_SZ];

    const int b   = blockIdx.y;
    const int i   = blockIdx.x;
    const int j   = threadIdx.x;
    const int tid = threadIdx.x;

    u_sh[j] = u[(size_t)b * K_SZ * E_SZ + (size_t)i * E_SZ + j];
    a_sh[j] = a_vec[j];

    const char* tchunk0 = (const char*)(tT + (size_t)b * E_SZ * K_SZ);
    const unsigned lds_base = (unsigned)(uintptr_t)t_sh;  // addr[31:0] == LDS offset

    float acc = 0.0f;
    for (int e0 = 0; e0 < E_SZ; e0 += CHUNK) {
        __syncthreads();  // previous chunk fully consumed (also covers u_sh/a_sh)
#if defined(__AMDGCN__)
        // Stage contiguous 32KB chunk tT[b, e0:e0+CHUNK, :] -> LDS.
        // 256 threads x 8 x 16B = 32KB, memory -> LDS without touching VGPRs.
#pragma unroll
        for (int inst = 0; inst < (CHUNK * K_SZ * 4) / (256 * 16); ++inst) {
            const unsigned off   = (unsigned)(inst * 4096 + tid * 16);
            const unsigned laddr = lds_base + off;
            const void*    gaddr = tchunk0 + (size_t)e0 * K_SZ * 4 + off;
            asm volatile("global_load_async_to_lds_b128 %0, %1, off"
                         :: "v"(laddr), "v"(gaddr)
                         : "memory");
        }
        asm volatile("s_wait_asynccnt 0x0" ::: "memory");
#else
        for (int inst = 0; inst < CHUNK * K_SZ / 256; ++inst)
            t_sh[inst * 256 + tid] =
                ((const float*)tchunk0)[(size_t)e0 * K_SZ + inst * 256 + tid];
#endif
        __syncthreads();

#pragma unroll 8
        for (int r = 0; r < CHUNK; ++r) {
            const float s = u_sh[e0 + r] + t_sh[r * K_SZ + j];
            acc = fmaf(a_sh[e0 + r], fmaxf(s, ALPHA * s), acc);
        }
    }
    const float ev = acc + bias[(size_t)i * K_SZ + j];

    // --- softmax over the 256 threads ---
    __syncthreads();
    red[j] = ev;
    __syncthreads();
    for (int s = K_SZ / 2; s > 0; s >>= 1) {
        if (j < s) red[j] = fmaxf(red[j], red[j + s]);
        __syncthreads();
    }
    const float m = red[0];
    __syncthreads();

    const float p = expf(ev - m);
    red[j] = p;
    __syncthreads();
    for (int s = K_SZ / 2; s > 0; s >>= 1) {
        if (j < s) red[j] += red[j + s];
        __syncthreads();
    }
    const float inv = 1.0f / red[0];

    attn[(size_t)b * K_SZ * K_SZ + (size_t)i * K_SZ + j] = p * inv;
}

// ---------------------------------------------------------------------------
// Kernel 3: h = sigmoid(attn @ v), stored transposed: out[b,w,i].
//   One wave per 16x16 tile, inner dim K=256 -> 64x V_WMMA_F32_16X16X4_F32.
// grid: (8 [w-tiles], 16 [i-tiles], 4 [batch])
// ---------------------------------------------------------------------------
__global__ void __launch_bounds__(32)
out_gemm_kernel(const float* __restrict__ attn,  // (B, K, K)
                const float* __restrict__ x,     // (B, W, K): v[b,j,w] = x[b,w,j]
                float* __restrict__ out)         // (B, W, K)
{
    const int lane = threadIdx.x;
    const int half = lane >> 4;
    const int lm   = lane & 15;

    const int w_base = blockIdx.x * 16;
    const int i_base = blockIdx.y * 16;
    const int b      = blockIdx.z;

    const float* ab = attn + (size_t)b * K_SZ * K_SZ;
    const float* xb = x + (size_t)b * W_SZ * K_SZ;

    v8f acc = {};
    for (int j0 = 0; j0 < K_SZ; j0 += 4) {
        const int kk = j0 + 2 * half;
        // A[m][kk] = attn[i_base+m, kk]
        const float* arow = ab + (size_t)(i_base + lm) * K_SZ + kk;
        v2f amat;
        amat.x = arow[0];
        amat.y = arow[1];
        // B[kk][n] = v[kk, w_base+n] = x[b, w_base+n, kk]
        const float* xrow = xb + (size_t)(w_base + lm) * K_SZ + kk;
        v2f bmat;
        bmat.x = xrow[0];
        bmat.y = xrow[1];
        acc = __builtin_amdgcn_wmma_f32_16x16x4_f32(
            false, amat, false, bmat, (short)0, acc, false, false);
    }

    // D tile element (m = i, n = w) -> out[b, w_base+n, i_base+m], sigmoid.
    float* op = out + (size_t)b * W_SZ * K_SZ + (size_t)(w_base + lm) * K_SZ + i_base;
#pragma unroll
    for (int r = 0; r < 8; ++r) {
        const int m = r + 8 * half;
        op[m] = 1.0f / (1.0f + expf(-acc[r]));
    }
}

// ---------------------------------------------------------------------------
extern "C" void kernel_launch(void* const* d_in, const int* in_sizes, int n_in,
                              void* d_out, int out_size, void* d_ws, size_t ws_size,
                              hipStream_t stream) {
    const float* x     = (const float*)d_in[0];   // (4,128,256)
    const float* lin_w = (const float*)d_in[1];   // (256,256)
    const float* lin_b = (const float*)d_in[2];   // (256)
    const float* a_vec = (const float*)d_in[3];   // (256)
    const float* bias  = (const float*)d_in[4];   // (256,256)
    float* out = (float*)d_out;                   // (4,128,256)

    float* u    = (float*)d_ws;                                   // B*K*E
    float* tT   = u  + (size_t)B_SZ * K_SZ * E_SZ;                // B*E*K
    float* attn = tT + (size_t)B_SZ * E_SZ * K_SZ;                // B*K*K

    dim3 g1(32, 16, B_SZ);
    gemm_ut_kernel<<<g1, 32, 0, stream>>>(x, lin_w, lin_b, u, tT);

    dim3 g2(K_SZ, B_SZ);
    attn_row_kernel<<<g2, 256, 0, stream>>>(u, tT, a_vec, bias, attn);

    dim3 g3(W_SZ / 16, K_SZ / 16, B_SZ);
    out_gemm_kernel<<<g3, 32, 0, stream>>>(attn, x, out);
}